// GAT_15487652069646
// MI455X (gfx1250) — compile-verified
//
#include <hip/hip_runtime.h>
#include <hip/hip_bf16.h>

// ---------------- constants from the reference ----------------
constexpr int kN  = 50000;   // nodes
constexpr int kE  = 800000;  // edges
constexpr int kD  = 128;     // hidden dim
constexpr int kG  = 500;     // graphs
constexpr int kNS = 100;     // num_stock
constexpr int kM0 = 64;      // mid_dim[0]
constexpr int kFC1 = kNS * kM0; // 6400
constexpr float kSlope = 0.2f;
constexpr float kBnEps = 1e-5f;

typedef __attribute__((ext_vector_type(2))) float v2f;
typedef __attribute__((ext_vector_type(8))) float v8f;

// ---------------- helpers ----------------
__device__ __forceinline__ void atomicMaxF(float* addr, float val) {
    // ordered-int encoding: correct for mixed signs, init = -inf
    if (val >= 0.0f) atomicMax((int*)addr, __float_as_int(val));
    else             atomicMin((unsigned int*)addr, __float_as_uint(val));
}

__device__ __forceinline__ float waveReduceAdd(float v) {
#pragma unroll
    for (int o = 16; o > 0; o >>= 1) v += __shfl_down(v, o, 32);
    return v;
}

// ---------------- GEMM: node transform (XL = X@WL+bL, XR = X@WR+bR) ----------------
// 512 threads = 16 waves. Waves 0..7 -> XL column tiles, waves 8..15 -> XR column
// tiles; one (matrix, col-tile) per wave keeps the hoisted B fragments at 64 VGPRs
// (no spill). B persists in registers across all row tiles (persistent blocks).
__global__ void transform_wmma(const float* __restrict__ X,
                               const float* __restrict__ WL, const float* __restrict__ bL,
                               const float* __restrict__ WR, const float* __restrict__ bR,
                               float* __restrict__ XL, float* __restrict__ XR,
                               int ntiles) {
    __shared__ float sA[16][kD];
    const int tid   = threadIdx.x;
    const int lane  = tid & 31;
    const int wave  = tid >> 5;          // 0..15
    const bool left = wave < 8;          // wave-uniform: EXEC stays all-ones
    const int ct    = left ? wave : wave - 8;
    const int n0    = ct * 16;
    const int lhalf = lane >> 4;         // 0 or 1
    const int l15   = lane & 15;
    const int n     = n0 + l15;

    const float* W    = left ? WL : WR;
    const float* bias = left ? bL : bR;
    float*       OUT  = left ? XL : XR;

    // hoist B fragments (K=128 -> 32 chunks of K=4) for this wave's matrix
    v2f fb[32];
#pragma unroll
    for (int kc = 0; kc < 32; ++kc) {
        int kk = kc * 4 + lhalf * 2;
        fb[kc].x = W[kk * kD + n];
        fb[kc].y = W[(kk + 1) * kD + n];
    }
    const float bv = bias[n];

    for (int t = blockIdx.x; t < ntiles; t += gridDim.x) {
        __syncthreads();
        for (int i = tid; i < 16 * kD; i += blockDim.x) {
            int r = i >> 7, c = i & 127;
            sA[r][c] = X[(size_t)(t * 16 + r) * kD + c];
        }
        __syncthreads();

        v8f acc = {};
#pragma unroll
        for (int kc = 0; kc < 32; ++kc) {
            int kk = kc * 4 + lhalf * 2;
            v2f a;
            a.x = sA[l15][kk];
            a.y = sA[l15][kk + 1];
            acc = __builtin_amdgcn_wmma_f32_16x16x4_f32(false, a, false, fb[kc],
                                                        (short)0, acc, false, false);
        }
#pragma unroll
        for (int r = 0; r < 8; ++r) {
            int m = r + lhalf * 8;
            size_t row = (size_t)t * 16 + m;
            OUT[row * kD + n] = acc[r] + bv;
        }
    }
}

// ---------------- per-layer init: zero accumulator, -inf max, zero denom ----------------
__global__ void init_layer(float* __restrict__ C, float* __restrict__ nmax,
                           float* __restrict__ den, int total, int n) {
    int i = blockIdx.x * blockDim.x + threadIdx.x;
    if (i < total) C[i] = 0.0f;
    if (i < n) { nmax[i] = -__builtin_inff(); den[i] = 0.0f; }
}

// ---------------- edge pass 1: attention logits + segment max ----------------
__global__ void edge_logit_max(const int* __restrict__ src, const int* __restrict__ dst,
                               const float* __restrict__ XL, const float* __restrict__ XR,
                               const float* __restrict__ att,
                               float* __restrict__ elog, float* __restrict__ nmax, int ne) {
    int wid  = (blockIdx.x * blockDim.x + threadIdx.x) >> 5;
    int lane = threadIdx.x & 31;
    if (wid >= ne) return;
    int s = src[wid], d = dst[wid];
    float4 a = ((const float4*)(XL + (size_t)s * kD))[lane];
    float4 b = ((const float4*)(XR + (size_t)d * kD))[lane];
    float4 w = ((const float4*)att)[lane];
    float acc = 0.0f, v;
    v = a.x + b.x; v = v > 0.0f ? v : kSlope * v; acc += w.x * v;
    v = a.y + b.y; v = v > 0.0f ? v : kSlope * v; acc += w.y * v;
    v = a.z + b.z; v = v > 0.0f ? v : kSlope * v; acc += w.z * v;
    v = a.w + b.w; v = v > 0.0f ? v : kSlope * v; acc += w.w * v;
    acc = waveReduceAdd(acc);
    if (lane == 0) { elog[wid] = acc; atomicMaxF(&nmax[d], acc); }
}

// ---------------- edge pass 2: exp + segment sum ----------------
__global__ void edge_softmax(const int* __restrict__ dst, float* __restrict__ elog,
                             const float* __restrict__ nmax, float* __restrict__ den, int ne) {
    int e = blockIdx.x * blockDim.x + threadIdx.x;
    if (e >= ne) return;
    int d = dst[e];
    float p = expf(elog[e] - nmax[d]);
    elog[e] = p;
    atomicAdd(&den[d], p);
}

// ---------------- edge pass 3: weighted scatter-add ----------------
__global__ void edge_aggregate(const int* __restrict__ src, const int* __restrict__ dst,
                               const float* __restrict__ XL, const float* __restrict__ elog,
                               const float* __restrict__ den, float* __restrict__ OUT, int ne) {
    int wid  = (blockIdx.x * blockDim.x + threadIdx.x) >> 5;
    int lane = threadIdx.x & 31;
    if (wid >= ne) return;
    int s = src[wid], d = dst[wid];
    float alpha = elog[wid] / den[d];
    float4 v = ((const float4*)(XL + (size_t)s * kD))[lane];
    float* o = OUT + (size_t)d * kD + lane * 4;
    atomicAdd(o + 0, alpha * v.x);
    atomicAdd(o + 1, alpha * v.y);
    atomicAdd(o + 2, alpha * v.z);
    atomicAdd(o + 3, alpha * v.w);
}

// ---------------- epilogue: +bias, BatchNorm(eval), ReLU (in place) ----------------
__global__ void bias_bn_relu(float* __restrict__ H, const float* __restrict__ b0,
                             const float* __restrict__ gamma, const float* __restrict__ beta,
                             const float* __restrict__ mean, const float* __restrict__ var,
                             int total) {
    int i = blockIdx.x * blockDim.x + threadIdx.x;
    if (i >= total) return;
    int c = i & (kD - 1);
    float v = H[i] + b0[c];
    v = gamma[c] * (v - mean[c]) * rsqrtf(var[c] + kBnEps) + beta[c];
    H[i] = v > 0.0f ? v : 0.0f;
}

__global__ void bias_add(float* __restrict__ H, const float* __restrict__ b, int total) {
    int i = blockIdx.x * blockDim.x + threadIdx.x;
    if (i >= total) return;
    H[i] += b[i & (kD - 1)];
}

// ---------------- global mean pool ----------------
__global__ void init_pool(float* __restrict__ pooled, float* __restrict__ cnt, int total, int g) {
    int i = blockIdx.x * blockDim.x + threadIdx.x;
    if (i < total) pooled[i] = 0.0f;
    if (i < g) cnt[i] = 0.0f;
}

__global__ void pool_accum(const float* __restrict__ H, const int* __restrict__ batch,
                           float* __restrict__ pooled, float* __restrict__ cnt, int total) {
    int i = blockIdx.x * blockDim.x + threadIdx.x;
    if (i >= total) return;
    int node = i >> 7, c = i & (kD - 1);
    int g = batch[node];
    atomicAdd(&pooled[(size_t)g * kD + c], H[i]);
    if (c == 0) atomicAdd(&cnt[g], 1.0f);
}

__global__ void pool_div(float* __restrict__ pooled, const float* __restrict__ cnt, int total) {
    int i = blockIdx.x * blockDim.x + threadIdx.x;
    if (i >= total) return;
    float c = cnt[i >> 7];
    pooled[i] /= (c > 1.0f ? c : 1.0f);
}

// ---------------- fc1: pooled[G,128] @ W[128,6400] + b, ReLU ----------------
__global__ void fc1_wmma(const float* __restrict__ P, const float* __restrict__ W,
                         const float* __restrict__ bias, float* __restrict__ OUT) {
    __shared__ float sA[16][kD];
    const int tid   = threadIdx.x;
    const int lane  = tid & 31;
    const int wave  = tid >> 5;
    const int lhalf = lane >> 4;
    const int l15   = lane & 15;
    const int row0  = blockIdx.x * 16;
    const int n0    = (blockIdx.y * 8 + wave) * 16;
    const int n     = n0 + l15;

    // prefetch all B fragments as one batch of outstanding loads
    v2f fb[32];
#pragma unroll
    for (int kc = 0; kc < 32; ++kc) {
        int kk = kc * 4 + lhalf * 2;
        fb[kc].x = W[(size_t)kk * kFC1 + n];
        fb[kc].y = W[(size_t)(kk + 1) * kFC1 + n];
    }

    for (int i = tid; i < 16 * kD; i += blockDim.x) {
        int r = i >> 7, c = i & 127;
        int row = row0 + r;
        sA[r][c] = (row < kG) ? P[(size_t)row * kD + c] : 0.0f;
    }
    __syncthreads();

    v8f acc = {};
#pragma unroll
    for (int kc = 0; kc < 32; ++kc) {
        int kk = kc * 4 + lhalf * 2;
        v2f a;
        a.x = sA[l15][kk];
        a.y = sA[l15][kk + 1];
        acc = __builtin_amdgcn_wmma_f32_16x16x4_f32(false, a, false, fb[kc],
                                                    (short)0, acc, false, false);
    }
    float bn = bias[n];
#pragma unroll
    for (int r = 0; r < 8; ++r) {
        int m = r + lhalf * 8, row = row0 + m;
        if (row < kG) {
            float v = acc[r] + bn;
            OUT[(size_t)row * kFC1 + n] = v > 0.0f ? v : 0.0f;
        }
    }
}

// ---------------- fc3 + sigmoid: one wave per (g, stock) output ----------------
__global__ void fc3_sigmoid(const float* __restrict__ H, const float* __restrict__ w,
                            const float* __restrict__ b, float* __restrict__ out, int total) {
    int wid  = (blockIdx.x * blockDim.x + threadIdx.x) >> 5;
    int lane = threadIdx.x & 31;
    if (wid >= total) return;
    float2 h  = ((const float2*)(H + (size_t)wid * kM0))[lane];
    float2 ww = ((const float2*)w)[lane];
    float acc = h.x * ww.x + h.y * ww.y;
    acc = waveReduceAdd(acc);
    if (lane == 0) out[wid] = 1.0f / (1.0f + expf(-(acc + b[0])));
}

// ---------------- host: launch sequence ----------------
extern "C" void kernel_launch(void* const* d_in, const int* in_sizes, int n_in,
                              void* d_out, int out_size, void* d_ws, size_t ws_size,
                              hipStream_t stream) {
    const float* x      = (const float*)d_in[0];
    const int*   gd     = (const int*)d_in[1];
    const int*   batch  = (const int*)d_in[2];
    const float* wl0    = (const float*)d_in[3];
    const float* bl0    = (const float*)d_in[4];
    const float* wr0    = (const float*)d_in[5];
    const float* br0    = (const float*)d_in[6];
    const float* att0   = (const float*)d_in[7];
    const float* b0     = (const float*)d_in[8];
    const float* wl1    = (const float*)d_in[9];
    const float* bl1    = (const float*)d_in[10];
    const float* wr1    = (const float*)d_in[11];
    const float* br1    = (const float*)d_in[12];
    const float* att1   = (const float*)d_in[13];
    const float* b1     = (const float*)d_in[14];
    const float* bng    = (const float*)d_in[15];
    const float* bnb    = (const float*)d_in[16];
    const float* bnm    = (const float*)d_in[17];
    const float* bnv    = (const float*)d_in[18];
    const float* fc1w   = (const float*)d_in[19];
    const float* fc1b   = (const float*)d_in[20];
    const float* fc3w   = (const float*)d_in[21];
    const float* fc3b   = (const float*)d_in[22];
    float* out = (float*)d_out;

    const int* src = gd;
    const int* dst = gd + kE;

    // workspace carve-out (floats)
    float* ws     = (float*)d_ws;
    float* XL     = ws;                               // N*D
    float* XR     = XL + (size_t)kN * kD;             // N*D
    float* C      = XR + (size_t)kN * kD;             // N*D (agg / h, reused both layers)
    float* ELOG   = C + (size_t)kN * kD;              // E
    float* NMAX   = ELOG + kE;                        // N
    float* DEN    = NMAX + kN;                        // N
    float* POOLED = DEN + kN;                         // G*D
    float* CNT    = POOLED + (size_t)kG * kD;         // G
    float* FC1O   = CNT + kG;                         // G * 6400

    const int ntiles   = kN / 16;                     // 3125
    const int nd       = kN * kD;                     // 6.4M
    const int bNDe     = (nd + 255) / 256;
    const int bEdgeW   = (kE * 32 + 255) / 256;       // one wave per edge
    const int bEdgeT   = (kE + 255) / 256;

    // ---- layer 0 ----
    transform_wmma<<<1024, 512, 0, stream>>>(x, wl0, bl0, wr0, br0, XL, XR, ntiles);
    init_layer<<<bNDe, 256, 0, stream>>>(C, NMAX, DEN, nd, kN);
    edge_logit_max<<<bEdgeW, 256, 0, stream>>>(src, dst, XL, XR, att0, ELOG, NMAX, kE);
    edge_softmax<<<bEdgeT, 256, 0, stream>>>(dst, ELOG, NMAX, DEN, kE);
    edge_aggregate<<<bEdgeW, 256, 0, stream>>>(src, dst, XL, ELOG, DEN, C, kE);
    bias_bn_relu<<<bNDe, 256, 0, stream>>>(C, b0, bng, bnb, bnm, bnv, nd);

    // ---- layer 1 ----
    transform_wmma<<<1024, 512, 0, stream>>>(C, wl1, bl1, wr1, br1, XL, XR, ntiles);
    init_layer<<<bNDe, 256, 0, stream>>>(C, NMAX, DEN, nd, kN);
    edge_logit_max<<<bEdgeW, 256, 0, stream>>>(src, dst, XL, XR, att1, ELOG, NMAX, kE);
    edge_softmax<<<bEdgeT, 256, 0, stream>>>(dst, ELOG, NMAX, DEN, kE);
    edge_aggregate<<<bEdgeW, 256, 0, stream>>>(src, dst, XL, ELOG, DEN, C, kE);
    bias_add<<<bNDe, 256, 0, stream>>>(C, b1, nd);

    // ---- global mean pool ----
    const int gdTot = kG * kD;
    init_pool<<<(gdTot + 255) / 256, 256, 0, stream>>>(POOLED, CNT, gdTot, kG);
    pool_accum<<<bNDe, 256, 0, stream>>>(C, batch, POOLED, CNT, nd);
    pool_div<<<(gdTot + 255) / 256, 256, 0, stream>>>(POOLED, CNT, gdTot);

    // ---- fc1 (WMMA) + ReLU ----
    dim3 g1((kG + 15) / 16, kFC1 / (8 * 16));         // (32, 50)
    fc1_wmma<<<g1, 256, 0, stream>>>(POOLED, fc1w, fc1b, FC1O);

    // ---- fc3 + sigmoid ----
    const int nout = kG * kNS;                        // 50000
    fc3_sigmoid<<<(nout * 32 + 255) / 256, 256, 0, stream>>>(FC1O, fc3w, fc3b, out, nout);
}